// CrfRnnLayerSPAT_3942779978353
// MI455X (gfx1250) — compile-verified
//
#include <hip/hip_runtime.h>
#include <cstdint>

// Problem constants (fixed by the reference)
#define C_       21
#define HW_      (1024 * 1024)
#define NSEG_    500
#define EPS_     1e-5f

// Tuning
#define TILE_            256                      // pixels staged per tile (per block)
#define ABLK_            256                      // blocks for the accumulate pass
#define TILES_TOTAL_     (HW_ / TILE_)            // 4096
#define TILES_PER_BLOCK_ (TILES_TOTAL_ / ABLK_)   // 16
#define FBLK_            1024                     // blocks for the finalize pass

// ---------------------------------------------------------------------------
// Pass 0: zero the (500 x 21) segment log-sum table in workspace.
// ---------------------------------------------------------------------------
__global__ void k_zero_seg(float* __restrict__ seg) {
    int i = blockIdx.x * blockDim.x + threadIdx.x;
    if (i < NSEG_ * C_) __builtin_nontemporal_store(0.0f, &seg[i]);
}

// ---------------------------------------------------------------------------
// Pass 1: per-pixel softmax -> log(q+eps), privatized segment-sum in LDS,
// global float-atomic flush. q_logits tiles are staged into LDS with the
// CDNA5 async global->LDS engine (ASYNCcnt), double buffered.
// Each lane async-loads exactly the 21 channel values of its own pixel, so a
// per-wave s_wait_asynccnt is sufficient (no cross-lane LDS consumption).
// q_logits is loaded with default (RT) temporal hints on purpose: it is
// re-read by pass 2 and fits in the 192 MB L2.
// ---------------------------------------------------------------------------
__global__ __launch_bounds__(256) void k_seg_accum(
    const float* __restrict__ q, const int* __restrict__ sp, float* __restrict__ seg)
{
    __shared__ float s_seg[NSEG_ * C_];        // 42000 B privatized histogram
    __shared__ float s_tile[2][C_ * TILE_];    // 2 x 21504 B staging buffers
    const int tid = threadIdx.x;

    for (int i = tid; i < NSEG_ * C_; i += 256) s_seg[i] = 0.0f;
    __syncthreads();

    const int t0 = blockIdx.x * TILES_PER_BLOCK_;

    auto issue_tile = [&](int tileIdx, int buf) {
        const int base = tileIdx * TILE_ + tid;   // pixel index for this lane
        #pragma unroll
        for (int r = 0; r < C_; ++r) {
            unsigned lds_off = (unsigned)(uintptr_t)&s_tile[buf][r * TILE_ + tid];
            unsigned g_off   = (unsigned)(((unsigned)r * (unsigned)HW_ + (unsigned)base) * 4u);
            // CDNA5 async DMA: copies 4 bytes from q plus byte offset g_off to the LDS address lds_off, tracked by ASYNCcnt
            asm volatile("global_load_async_to_lds_b32 %0, %1, %2"
                         :: "v"(lds_off), "v"(g_off), "s"(q) : "memory");
        }
    };

    issue_tile(t0, 0);
    for (int t = 0; t < TILES_PER_BLOCK_; ++t) {
        const int buf = t & 1;
        if (t + 1 < TILES_PER_BLOCK_) {
            issue_tile(t0 + t + 1, buf ^ 1);                  // prefetch next tile
            asm volatile("s_wait_asynccnt 21" ::: "memory");  // current tile (21 loads) done
        } else {
            asm volatile("s_wait_asynccnt 0" ::: "memory");
        }

        // Softmax over the 21 channels of this lane's pixel.
        float v[C_];
        float m = -__builtin_inff();
        #pragma unroll
        for (int r = 0; r < C_; ++r) {
            v[r] = s_tile[buf][r * TILE_ + tid];
            m = fmaxf(m, v[r]);
        }
        float ssum = 0.0f;
        #pragma unroll
        for (int r = 0; r < C_; ++r) { v[r] = __expf(v[r] - m); ssum += v[r]; }
        const float inv = __builtin_amdgcn_rcpf(ssum);

        const int pix = (t0 + t) * TILE_ + tid;
        const int sgi = sp[pix];
        float* dst = &s_seg[sgi * C_];
        #pragma unroll
        for (int r = 0; r < C_; ++r) {
            atomicAdd(&dst[r], __logf(v[r] * inv + EPS_));    // ds_add_f32
        }
    }
    __syncthreads();

    // Cross-block reduction: native global_atomic_add_f32 into the 42 KB table.
    for (int i = tid; i < NSEG_ * C_; i += 256) {
        unsafeAtomicAdd(&seg[i], s_seg[i]);
    }
}

// ---------------------------------------------------------------------------
// Pass 2: recompute softmax (cheaper than an 84 MB logq round-trip; q_logits
// should still be L2-resident from pass 1), gather the finished segment table
// from LDS, and emit the fused superpixel + attachment update.
// Output stores are non-temporal (write-once stream) so the 84 MB of stores
// do not evict q_logits from L2.
// ---------------------------------------------------------------------------
__global__ __launch_bounds__(256) void k_finalize(
    const float* __restrict__ q, const float* __restrict__ lw, const float* __restrict__ hw,
    const int* __restrict__ sp, const float* __restrict__ seg, float* __restrict__ out)
{
    __shared__ float s_seg[NSEG_ * C_];
    __shared__ float s_lw[2 * C_];
    __shared__ float s_hw[2];
    const int tid = threadIdx.x;
    for (int i = tid; i < NSEG_ * C_; i += 256) s_seg[i] = seg[i];
    if (tid < 2 * C_) s_lw[tid] = lw[tid];
    if (tid < 2)      s_hw[tid] = hw[tid];
    __syncthreads();

    const float hw0 = s_hw[0], hw1 = s_hw[1];
    const int stride = gridDim.x * blockDim.x;
    for (int p = blockIdx.x * blockDim.x + tid; p < HW_; p += stride) {
        float v[C_];
        float m = -__builtin_inff();
        #pragma unroll
        for (int r = 0; r < C_; ++r) { v[r] = q[r * HW_ + p]; m = fmaxf(m, v[r]); }
        float ssum = 0.0f;
        #pragma unroll
        for (int r = 0; r < C_; ++r) { v[r] = __expf(v[r] - m); ssum += v[r]; }
        const float inv = __builtin_amdgcn_rcpf(ssum);

        const int sgi = sp[p];
        const float* B = &s_seg[sgi * C_];
        #pragma unroll
        for (int r = 0; r < C_; ++r) {
            const float qv   = v[r] * inv;
            const float psp  = B[r];                        // prod_sp
            const float patt = (float)(NSEG_ - 1) * psp;    // prod_att
            const float qm   = (qv == 0.0f) ? 1.0f : qv;    // q + (q==0)
            const float den  = __logf(qm + EPS_);
            const float fsp  = __expf(psp  - den);
            const float fatt = __expf(patt - den);
            const float o = s_lw[r]      * fsp  + hw0 * (1.0f - fsp)
                          + s_lw[C_ + r] * fatt + hw1 * (1.0f - fatt);
            __builtin_nontemporal_store(o, &out[r * HW_ + p]);
        }
    }
}

// ---------------------------------------------------------------------------
extern "C" void kernel_launch(void* const* d_in, const int* in_sizes, int n_in,
                              void* d_out, int out_size, void* d_ws, size_t ws_size,
                              hipStream_t stream) {
    (void)in_sizes; (void)n_in; (void)out_size; (void)ws_size;
    const float* q   = (const float*)d_in[0];   // (21, 1024, 1024) f32
    const float* lw  = (const float*)d_in[1];   // (2, 21) f32
    const float* hw  = (const float*)d_in[2];   // (2,) f32
    const int*   sp  = (const int*)d_in[3];     // (1024, 1024) i32
    float* out = (float*)d_out;                 // (21, 1024, 1024) f32
    float* seg = (float*)d_ws;                  // (500, 21) f32 scratch

    k_zero_seg<<<(NSEG_ * C_ + 255) / 256, 256, 0, stream>>>(seg);
    k_seg_accum<<<ABLK_, 256, 0, stream>>>(q, sp, seg);
    k_finalize<<<FBLK_, 256, 0, stream>>>(q, lw, hw, sp, seg, out);
}